// IndexedMaxPool2d_13219909337238
// MI455X (gfx1250) — compile-verified
//
#include <hip/hip_runtime.h>
#include <hip/hip_bf16.h>

#define B_ 16
#define F_ 256
#define N_ 16384
#define L_ 4096
#define K_ 9

#define THREADS 256

// ---- CDNA5 async global->LDS path (gfx1250), guarded so compile never breaks ----
#if defined(__has_builtin)
#  if __has_builtin(__builtin_amdgcn_global_load_async_to_lds_b128)
#    define HAVE_ASYNC_LDS 1
#  endif
#  if __has_builtin(__builtin_amdgcn_s_wait_asynccnt)
#    define HAVE_WAIT_ASYNC 1
#  endif
#endif

// GCC-style vector type matching the builtin's parameter type:
//   'int __vector(4) __device__ *'  ==  v4i32 addrspace(1)*
typedef int v4i __attribute__((vector_size(16)));
typedef __attribute__((address_space(1))) v4i* gptr4_t;  // global
typedef __attribute__((address_space(3))) v4i* lptr4_t;  // LDS

__device__ __forceinline__ void wait_async_zero() {
#if defined(HAVE_WAIT_ASYNC)
    __builtin_amdgcn_s_wait_asynccnt(0);
#elif defined(HAVE_ASYNC_LDS)
    asm volatile("s_wait_asynccnt 0" ::: "memory");
#endif
}

// Transposed pack: packedT[k*L + l] = (idx[l,k] << 2) | (mask[l,k]!=0 ? bit31 : 0).
// Byte offset (idx<<2) <= 0xFFFC fits in low 16 bits; flag in bit 31.
__global__ __launch_bounds__(THREADS) void pack_idx_mask_kernel(
    const int* __restrict__ idx, const float* __restrict__ mask,
    unsigned* __restrict__ packedT, int n)
{
    int i = blockIdx.x * THREADS + threadIdx.x;
    if (i < n) {
        int l = i / K_;
        int k = i - l * K_;
        unsigned p = ((unsigned)idx[i] << 2) & 0xFFFFu;
        if (mask[i] != 0.0f) p |= 0x80000000u;
        packedT[(size_t)k * L_ + l] = p;
    }
}

__device__ __forceinline__ float masked_max9(const char* planeb, const unsigned* p) {
    // k=0 seeds the max; entries with clear flag contribute 0.0f (== val*mask)
    float m = (p[0] & 0x80000000u)
                  ? *(const float*)(planeb + (p[0] & 0xFFFFu)) : 0.0f;
    #pragma unroll
    for (int k = 1; k < K_; ++k) {
        float v = *(const float*)(planeb + (p[k] & 0xFFFFu)); // ds_load_b32 gather
        v = (p[k] & 0x80000000u) ? v : 0.0f;                  // v_cndmask
        m = fmaxf(m, v);                                      // v_max_num_f32
    }
    return m;
}

// One block per (b,f) plane. Plane (64KB) is DMA'd into LDS via async loads,
// then each thread max-pools over K=9 gathered neighbors for L/THREADS pixels.
__global__ __launch_bounds__(THREADS) void indexed_maxpool_kernel(
    const float* __restrict__ input,      // (B*F, N)
    const unsigned* __restrict__ packedT, // (K, L) byteoff|maskbit
    float* __restrict__ out)              // (B*F, L)
{
    __shared__ float plane[N_];           // 64 KB -> 4 blocks / WGP

    const int tid = threadIdx.x;
    const int bf  = blockIdx.x;
    const float* __restrict__ src = input + (size_t)bf * N_;

    // ---- stage 1: plane -> LDS (async DMA, 16 B per lane per issue) ----
#if defined(HAVE_ASYNC_LDS)
    #pragma unroll
    for (int i = tid * 4; i < N_; i += THREADS * 4) {
        __builtin_amdgcn_global_load_async_to_lds_b128(
            (gptr4_t)(src + i), (lptr4_t)(plane + i), 0, 0);
    }
#else
    #pragma unroll
    for (int i = tid * 4; i < N_; i += THREADS * 4) {
        *(float4*)(plane + i) = *(const float4*)(src + i);
    }
#endif

    // Preload iteration-0 indices while the DMA is in flight (one addr VGPR,
    // k*L_*4 becomes an immediate offset on each global_load_b32).
    unsigned p0[K_];
    {
        const unsigned* __restrict__ q = packedT + tid;
        #pragma unroll
        for (int k = 0; k < K_; ++k) p0[k] = q[(size_t)k * L_];
    }

#if defined(HAVE_ASYNC_LDS)
    wait_async_zero();
#endif
    __syncthreads();

    // ---- stage 2: gather + masked max over K ----
    const char* planeb = (const char*)plane;
    float* __restrict__ outp = out + (size_t)bf * L_;

    outp[tid] = masked_max9(planeb, p0);

    for (int l = tid + THREADS; l < L_; l += THREADS) {
        unsigned p[K_];
        const unsigned* __restrict__ q = packedT + l;
        #pragma unroll
        for (int k = 0; k < K_; ++k) p[k] = q[(size_t)k * L_];
        outp[l] = masked_max9(planeb, p);
    }
}

// Fallback if workspace is too small for the packed table: read idx+mask raw.
__global__ __launch_bounds__(THREADS) void indexed_maxpool_raw_kernel(
    const float* __restrict__ input, const int* __restrict__ idx,
    const float* __restrict__ mask, float* __restrict__ out)
{
    __shared__ float plane[N_];
    const int tid = threadIdx.x;
    const int bf  = blockIdx.x;
    const float* __restrict__ src = input + (size_t)bf * N_;

#if defined(HAVE_ASYNC_LDS)
    #pragma unroll
    for (int i = tid * 4; i < N_; i += THREADS * 4) {
        __builtin_amdgcn_global_load_async_to_lds_b128(
            (gptr4_t)(src + i), (lptr4_t)(plane + i), 0, 0);
    }
    wait_async_zero();
#else
    #pragma unroll
    for (int i = tid * 4; i < N_; i += THREADS * 4) {
        *(float4*)(plane + i) = *(const float4*)(src + i);
    }
#endif
    __syncthreads();

    float* __restrict__ outp = out + (size_t)bf * L_;
    for (int l = tid; l < L_; l += THREADS) {
        const int*   __restrict__ ip = idx  + (size_t)l * K_;
        const float* __restrict__ mp = mask + (size_t)l * K_;
        float m = -__builtin_huge_valf();
        #pragma unroll
        for (int k = 0; k < K_; ++k) {
            float v = plane[ip[k]] * mp[k];
            m = fmaxf(m, v);
        }
        outp[l] = m;
    }
}

extern "C" void kernel_launch(void* const* d_in, const int* in_sizes, int n_in,
                              void* d_out, int out_size, void* d_ws, size_t ws_size,
                              hipStream_t stream) {
    const float* input = (const float*)d_in[0];   // (B,F,N) fp32
    const int*   idx   = (const int*)  d_in[1];   // (L,K) int32
    const float* mask  = (const float*)d_in[2];   // (L,K) fp32
    float*       out   = (float*)d_out;           // (B,F,L) fp32

    const int LK = L_ * K_;
    const int nblocks = B_ * F_;                  // 4096 planes

    if (ws_size >= (size_t)LK * sizeof(unsigned)) {
        unsigned* packedT = (unsigned*)d_ws;
        pack_idx_mask_kernel<<<(LK + THREADS - 1) / THREADS, THREADS, 0, stream>>>(
            idx, mask, packedT, LK);
        indexed_maxpool_kernel<<<nblocks, THREADS, 0, stream>>>(input, packedT, out);
    } else {
        indexed_maxpool_raw_kernel<<<nblocks, THREADS, 0, stream>>>(input, idx, mask, out);
    }
}